// HyperGsysHGNN_27831388078171
// MI455X (gfx1250) — compile-verified
//
#include <hip/hip_runtime.h>
#include <hip/hip_bf16.h>

typedef float v2f_ __attribute__((ext_vector_type(2)));
typedef float v8f_ __attribute__((ext_vector_type(8)));

#define NV_   50000
#define NE_   25000
#define NNZ_  400000
#define DIM_  512

// ---------------------------------------------------------------------------
// Zero a buffer (float4 granularity). Sizes here are all multiples of 4.
// ---------------------------------------------------------------------------
__global__ __launch_bounds__(256) void hg_zero4(float4* __restrict__ p, int n4) {
    int i = blockIdx.x * blockDim.x + threadIdx.x;
    if (i < n4) p[i] = make_float4(0.f, 0.f, 0.f, 0.f);
}

// ---------------------------------------------------------------------------
// Xp[n][o] = sum_d X[n][d] * W[o][d]   (nn.Linear, no bias)
//
// FP32 WMMA 16x16x4 with 2-D register blocking: each wave owns a 64x64 output
// tile = 4x4 grid of 16x16 WMMA tiles (16 accumulators, 128 VGPRs). Per
// k-step: 4 A-fragment + 4 B-fragment b64 loads feed 16 WMMAs (0.5 loads per
// WMMA vs 1.25 in the naive strip version). Block = 8 waves (2 row x 4 col)
// covering 128x256; grid (ceil(N/128), 512/256).
//
// Fragment layouts (ISA 7.12.2, wave32):
//   A 16x4 : lane l (m=l&15, kb=2*(l>>4)) holds {X[m][kb], X[m][kb+1]}
//   B 4x16 : lane l (n=l&15, kb=2*(l>>4)) holds {B[kb][n], B[kb+1][n]},
//            B[k][n] = W[col+n][k]
//   C 16x16: vgpr v, lane l -> row v + 8*(l>>4), col l&15
//
// Ragged edge: NV_ % 16 == 0, so every 16-row sub-tile is either fully valid
// or fully out of range. OOB sub-tiles: loads clamped in-bounds, stores
// skipped (condition is uniform per sub-tile -> no divergence).
// ---------------------------------------------------------------------------
__global__ __launch_bounds__(256) void hg_gemm_wmma(const float* __restrict__ X,
                                                    const float* __restrict__ Wt,
                                                    float* __restrict__ Xp) {
    const int lane = threadIdx.x & 31;
    const int wv   = threadIdx.x >> 5;
    const int wrow = wv >> 2;                       // 0..1
    const int wcol = wv & 3;                        // 0..3

    const int row0 = blockIdx.x * 128 + wrow * 64;  // <= 49984
    const int col0 = blockIdx.y * 256 + wcol * 64;

    const int m  = lane & 15;
    const int kb = (lane >> 4) << 1;                // 0 or 2

    const float* ap[4];
#pragma unroll
    for (int r = 0; r < 4; ++r) {
        int rr = row0 + r * 16;
        if (rr >= NV_) rr = NV_ - 16;               // clamp: reads stay in-bounds
        ap[r] = X + (size_t)(rr + m) * DIM_ + kb;
    }
    const float* bp = Wt + (size_t)(col0 + m) * DIM_ + kb;

    v8f_ c[4][4];
#pragma unroll
    for (int r = 0; r < 4; ++r)
#pragma unroll
        for (int q = 0; q < 4; ++q)
            c[r][q] = (v8f_){};

#pragma unroll 2
    for (int k = 0; k < DIM_; k += 4) {
        v2f_ a[4], b[4];
#pragma unroll
        for (int r = 0; r < 4; ++r)
            a[r] = *(const v2f_*)(ap[r] + k);
#pragma unroll
        for (int q = 0; q < 4; ++q)
            b[q] = *(const v2f_*)(bp + (size_t)q * 16 * DIM_ + k);  // ioffset-folded
#pragma unroll
        for (int r = 0; r < 4; ++r)
#pragma unroll
            for (int q = 0; q < 4; ++q)
                c[r][q] = __builtin_amdgcn_wmma_f32_16x16x4_f32(
                    false, a[r], false, b[q], (short)0, c[r][q], false, false);
    }

    const int rb = (lane >> 4) << 3;                // +8 rows for upper half-wave
    const int nc = lane & 15;
#pragma unroll
    for (int r = 0; r < 4; ++r) {
        const int rowt = row0 + r * 16;
        if (rowt >= NV_) continue;                  // uniform: whole sub-tile OOB
        float* o0 = Xp + (size_t)(rowt + rb) * DIM_ + col0 + nc;
#pragma unroll
        for (int v = 0; v < 8; ++v) {
            float* orow = o0 + (size_t)v * DIM_;
#pragma unroll
            for (int q = 0; q < 4; ++q)
                orow[q * 16] = c[r][q][v];
        }
    }
}

// ---------------------------------------------------------------------------
// Stage 1: Xe[e] += Xp[v] over incidence pairs. One wave per pair; b128
// gathers (HBM), f32 atomics land in L2 (Xe = 51.2 MB fits the 192 MB L2).
// ---------------------------------------------------------------------------
__global__ __launch_bounds__(256) void hg_scatter_v2e(const float* __restrict__ Xp,
                                                      const int* __restrict__ vidx,
                                                      const int* __restrict__ eidx,
                                                      float* __restrict__ Xe) {
    const int pair = (blockIdx.x << 3) + (threadIdx.x >> 5);
    const int lane = threadIdx.x & 31;
    if (pair >= NNZ_) return;
    const int v = vidx[pair];
    const int e = eidx[pair];
    const float4* src = (const float4*)(Xp + (size_t)v * DIM_);
    float* dst = Xe + (size_t)e * DIM_;
#pragma unroll
    for (int j = 0; j < 4; ++j) {
        const int q = lane + (j << 5);              // float4 index 0..127
        float4 x = src[q];
        atomicAdd(dst + (q << 2) + 0, x.x);
        atomicAdd(dst + (q << 2) + 1, x.y);
        atomicAdd(dst + (q << 2) + 2, x.z);
        atomicAdd(dst + (q << 2) + 3, x.w);
    }
}

// ---------------------------------------------------------------------------
// Stage 2: out[v] += Xe[e] * (degE[e]*Wdiag[e]). Xe gathers hit L2.
// ---------------------------------------------------------------------------
__global__ __launch_bounds__(256) void hg_scatter_e2v(const float* __restrict__ Xe,
                                                      const float* __restrict__ degE,
                                                      const float* __restrict__ Wdiag,
                                                      const int* __restrict__ vidx,
                                                      const int* __restrict__ eidx,
                                                      float* __restrict__ out) {
    const int pair = (blockIdx.x << 3) + (threadIdx.x >> 5);
    const int lane = threadIdx.x & 31;
    if (pair >= NNZ_) return;
    const int v = vidx[pair];
    const int e = eidx[pair];
    const float s = degE[e] * Wdiag[e];
    const float4* src = (const float4*)(Xe + (size_t)e * DIM_);
    float* dst = out + (size_t)v * DIM_;
#pragma unroll
    for (int j = 0; j < 4; ++j) {
        const int q = lane + (j << 5);
        float4 x = src[q];
        atomicAdd(dst + (q << 2) + 0, x.x * s);
        atomicAdd(dst + (q << 2) + 1, x.y * s);
        atomicAdd(dst + (q << 2) + 2, x.z * s);
        atomicAdd(dst + (q << 2) + 3, x.w * s);
    }
}

// ---------------------------------------------------------------------------
// Final scale: out[n][:] *= degV[n]
// ---------------------------------------------------------------------------
__global__ __launch_bounds__(256) void hg_scale_degv(float4* __restrict__ out,
                                                     const float* __restrict__ degV) {
    int i = blockIdx.x * blockDim.x + threadIdx.x;  // float4 index
    if (i >= NV_ * (DIM_ / 4)) return;
    const int row = i >> 7;                          // DIM_/4 = 128
    const float s = degV[row];
    float4 x = out[i];
    x.x *= s; x.y *= s; x.z *= s; x.w *= s;
    out[i] = x;
}

extern "C" void kernel_launch(void* const* d_in, const int* in_sizes, int n_in,
                              void* d_out, int out_size, void* d_ws, size_t ws_size,
                              hipStream_t stream) {
    (void)in_sizes; (void)n_in; (void)out_size; (void)ws_size;
    const float* X     = (const float*)d_in[0];
    const float* W     = (const float*)d_in[1];
    const float* degE  = (const float*)d_in[2];
    const float* degV  = (const float*)d_in[3];
    const float* Wdiag = (const float*)d_in[4];
    const int*   vidx  = (const int*)d_in[5];
    const int*   eidx  = (const int*)d_in[6];
    float* out = (float*)d_out;

    // Workspace layout: Xp [NV_ x DIM_] then Xe [NE_ x DIM_]  (~153.6 MB)
    float* Xp = (float*)d_ws;
    float* Xe = Xp + (size_t)NV_ * DIM_;

    const int ne4 = NE_ * (DIM_ / 4);
    const int nv4 = NV_ * (DIM_ / 4);

    hg_zero4<<<(ne4 + 255) / 256, 256, 0, stream>>>((float4*)Xe, ne4);
    hg_zero4<<<(nv4 + 255) / 256, 256, 0, stream>>>((float4*)out, nv4);

    dim3 ggrid((NV_ + 127) / 128, DIM_ / 256);       // (391, 2)
    hg_gemm_wmma<<<ggrid, 256, 0, stream>>>(X, W, Xp);

    hg_scatter_v2e<<<NNZ_ / 8, 256, 0, stream>>>(Xp, vidx, eidx, Xe);
    hg_scatter_e2v<<<NNZ_ / 8, 256, 0, stream>>>(Xe, degE, Wdiag, vidx, eidx, out);

    hg_scale_degv<<<(nv4 + 255) / 256, 256, 0, stream>>>((float4*)out, degV);
}